// GRU_18829136625956
// MI455X (gfx1250) — compile-verified
//
#include <hip/hip_runtime.h>

typedef float v2f __attribute__((ext_vector_type(2)));
typedef float v8f __attribute__((ext_vector_type(8)));

#define T_SEQ 65536
#define NINP  512
#define HID   8
#define G     24      // 3*H, gate rows (r,z,n)
#define NOUT  32
#define CHUNK 256     // scan pipeline chunk (steps)
#define WSTR  516     // LDS row stride for w_ih0 (516%64=4 -> conflict-free)

// ---------------------------------------------------------------------------
// Kernel 1: gi0 = x @ w_ih0^T + b_ih0   (M=T, K=512, N=24 padded to 32)
// f32 WMMA 16x16x4. Each wave owns a 16-row M tile and both 16-col N tiles.
// A frag (16x4 f32): lane<16 -> {x[m0+lr][k0], [k0+1]}, lane>=16 -> k0+2,k0+3.
// B frag (4x16 f32): mirrored lane-half K split, N striped across lanes.
// ---------------------------------------------------------------------------
__global__ __launch_bounds__(256) void gi0_wmma(const float* __restrict__ x,
                                                const float* __restrict__ w,    // [24,512]
                                                const float* __restrict__ bias, // [24]
                                                float* __restrict__ gi)         // [T,24]
{
  __shared__ float wlds[32 * WSTR];
  const int tid = threadIdx.x;
  for (int i = tid; i < 32 * NINP; i += 256) {
    int n = i >> 9, k = i & 511;
    wlds[n * WSTR + k] = (n < G) ? w[n * NINP + k] : 0.0f;
  }
  __syncthreads();

  const int lane = tid & 31, wave = tid >> 5;
  const int half = lane >> 4, lr = lane & 15;
  const int m0 = (blockIdx.x * 8 + wave) * 16;

  const float* arow = x + (size_t)(m0 + lr) * NINP + 2 * half;
  const float* b0p  = &wlds[lr * WSTR + 2 * half];
  const float* b1p  = &wlds[(16 + lr) * WSTR + 2 * half];

  v8f acc0 = {}; v8f acc1 = {};
#pragma unroll 4
  for (int k0 = 0; k0 < NINP; k0 += 4) {
    v2f a  = *(const v2f*)(arow + k0);
    v2f b0 = *(const v2f*)(b0p + k0);
    v2f b1 = *(const v2f*)(b1p + k0);
    acc0 = __builtin_amdgcn_wmma_f32_16x16x4_f32(false, a, false, b0, (short)0, acc0, false, false);
    acc1 = __builtin_amdgcn_wmma_f32_16x16x4_f32(false, a, false, b1, (short)0, acc1, false, false);
  }

  const float bia0 = bias[lr];
  const float bia1 = (lr < 8) ? bias[16 + lr] : 0.0f;
#pragma unroll
  for (int i = 0; i < 8; ++i) {
    int row = m0 + i + 8 * half;                    // C/D layout: vgpr i -> M=i (+8 for hi half)
    gi[(size_t)row * G + lr] = acc0[i] + bia0;
    if (lr < 8) gi[(size_t)row * G + 16 + lr] = acc1[i] + bia1;
  }
}

// ---------------------------------------------------------------------------
// Kernel 2: pipelined 3-layer GRU scan. One workgroup, 3 wave32 (one/layer),
// chunked handoff through LDS double buffers, barrier per phase.
// Lane r<24 owns gate row r; gates combined via wave32 shuffles.
// ---------------------------------------------------------------------------
__device__ __forceinline__ float sigm(float xv) {
  return __builtin_amdgcn_rcpf(1.0f + __expf(-xv));
}
__device__ __forceinline__ float fast_tanh(float xv) {
  float e = __expf(-2.0f * xv);
  return (1.0f - e) * __builtin_amdgcn_rcpf(1.0f + e);
}

__global__ __launch_bounds__(96) void gru_scan3(
    const float* __restrict__ gi0,   // [T,24] (already includes b_ih0)
    const float* __restrict__ whh0, const float* __restrict__ bhh0,
    const float* __restrict__ wih1, const float* __restrict__ whh1,
    const float* __restrict__ bih1, const float* __restrict__ bhh1,
    const float* __restrict__ wih2, const float* __restrict__ whh2,
    const float* __restrict__ bih2, const float* __restrict__ bhh2,
    float* __restrict__ h2out)       // [T,8]
{
  __shared__ float h01[2][CHUNK][8];   // layer0 -> layer1
  __shared__ float h12[2][CHUNK][8];   // layer1 -> layer2

  const int lane = threadIdx.x & 31;
  const int wv   = threadIdx.x >> 5;   // 0,1,2 = layer id

  const float* whh = (wv == 0) ? whh0 : (wv == 1) ? whh1 : whh2;
  const float* bhhp= (wv == 0) ? bhh0 : (wv == 1) ? bhh1 : bhh2;
  const float* wih = (wv == 1) ? wih1 : wih2;
  const float* bih = (wv == 1) ? bih1 : bih2;

  float wh[8], wi[8];
  float bh = 0.0f, bi = 0.0f;
#pragma unroll
  for (int j = 0; j < 8; ++j) { wh[j] = 0.0f; wi[j] = 0.0f; }
  if (lane < G) {
#pragma unroll
    for (int j = 0; j < 8; ++j) wh[j] = whh[lane * 8 + j];
    bh = bhhp[lane];
    if (wv > 0) {
#pragma unroll
      for (int j = 0; j < 8; ++j) wi[j] = wih[lane * 8 + j];
      bi = bih[lane];
    }
  }

  // replicated hidden state (all lanes) + own-lane h (lanes 0..7)
  float hr0=0.f,hr1=0.f,hr2=0.f,hr3=0.f,hr4=0.f,hr5=0.f,hr6=0.f,hr7=0.f;
  float hl = 0.f;

  const int NC = T_SEQ / CHUNK;
  for (int p = 0; p < NC + 2; ++p) {
    const int c = p - wv;
    if (c >= 0 && c < NC) {
      const int buf = c & 1;
      const int t0  = c * CHUNK;
#pragma unroll 2
      for (int tt = 0; tt < CHUNK; ++tt) {
        const int t = t0 + tt;
        float gi;
        if (wv == 0) {
          gi = (lane < G) ? gi0[(size_t)t * G + lane] : 0.0f;
        } else {
          const float* src = (wv == 1) ? &h01[buf][tt][0] : &h12[buf][tt][0];
          float4 xa = *(const float4*)(src);
          float4 xb = *(const float4*)(src + 4);
          float g01 = fmaf(wi[0], xa.x, wi[1] * xa.y);
          float g23 = fmaf(wi[2], xa.z, wi[3] * xa.w);
          float g45 = fmaf(wi[4], xb.x, wi[5] * xb.y);
          float g67 = fmaf(wi[6], xb.z, wi[7] * xb.w);
          gi = (g01 + g23) + (g45 + g67) + bi;
        }
        // gh = w_hh[row] . h + b_hh[row]   (balanced FMA tree)
        float s01 = fmaf(wh[0], hr0, wh[1] * hr1);
        float s23 = fmaf(wh[2], hr2, wh[3] * hr3);
        float s45 = fmaf(wh[4], hr4, wh[5] * hr5);
        float s67 = fmaf(wh[6], hr6, wh[7] * hr7);
        float gh  = (s01 + s23) + (s45 + s67) + bh;
        float ss  = gi + gh;
        // lanes 0..7 combine the three gate rows (own, +8, +16)
        float r   = sigm(ss);
        float sz  = __shfl(ss, lane + 8, 32);
        float z   = sigm(sz);
        float gin = __shfl(gi, lane + 16, 32);
        float ghn = __shfl(gh, lane + 16, 32);
        float n   = fast_tanh(fmaf(r, ghn, gin));
        float hn  = fmaf(z, hl - n, n);        // (1-z)*n + z*h
        hl = hn;
        // broadcast new hidden state to all lanes
        hr0 = __shfl(hn, 0, 32); hr1 = __shfl(hn, 1, 32);
        hr2 = __shfl(hn, 2, 32); hr3 = __shfl(hn, 3, 32);
        hr4 = __shfl(hn, 4, 32); hr5 = __shfl(hn, 5, 32);
        hr6 = __shfl(hn, 6, 32); hr7 = __shfl(hn, 7, 32);
        if (lane < 8) {
          if      (wv == 0) h01[buf][tt][lane] = hn;
          else if (wv == 1) h12[buf][tt][lane] = hn;
          else              h2out[(size_t)t * HID + lane] = hn;
        }
      }
    }
    __syncthreads();
  }
}

// ---------------------------------------------------------------------------
// Kernel 3: out = relu(h2) @ w_lin^T + b_lin   (M=T, K=8, N=32) via f32 WMMA.
// ---------------------------------------------------------------------------
__global__ __launch_bounds__(256) void out_wmma(const float* __restrict__ h2,   // [T,8]
                                                const float* __restrict__ wlin, // [32,8]
                                                const float* __restrict__ blin, // [32]
                                                float* __restrict__ out)        // [T,32]
{
  const int tid = threadIdx.x, lane = tid & 31, wave = tid >> 5;
  const int half = lane >> 4, lr = lane & 15;
  const int m0 = (blockIdx.x * 8 + wave) * 16;

  const float* arow = h2 + (size_t)(m0 + lr) * HID + 2 * half;
  v8f acc0 = {}; v8f acc1 = {};
#pragma unroll
  for (int k0 = 0; k0 < HID; k0 += 4) {
    v2f a = *(const v2f*)(arow + k0);
    a.x = fmaxf(a.x, 0.0f); a.y = fmaxf(a.y, 0.0f);      // fused ReLU
    v2f b0 = *(const v2f*)(wlin + lr * HID + k0 + 2 * half);
    v2f b1 = *(const v2f*)(wlin + (16 + lr) * HID + k0 + 2 * half);
    acc0 = __builtin_amdgcn_wmma_f32_16x16x4_f32(false, a, false, b0, (short)0, acc0, false, false);
    acc1 = __builtin_amdgcn_wmma_f32_16x16x4_f32(false, a, false, b1, (short)0, acc1, false, false);
  }
  const float bb0 = blin[lr], bb1 = blin[16 + lr];
#pragma unroll
  for (int i = 0; i < 8; ++i) {
    int row = m0 + i + 8 * half;
    out[(size_t)row * NOUT + lr]      = acc0[i] + bb0;
    out[(size_t)row * NOUT + 16 + lr] = acc1[i] + bb1;
  }
}

// ---------------------------------------------------------------------------
extern "C" void kernel_launch(void* const* d_in, const int* in_sizes, int n_in,
                              void* d_out, int out_size, void* d_ws, size_t ws_size,
                              hipStream_t stream) {
  const float* x    = (const float*)d_in[0];
  const float* wih0 = (const float*)d_in[1];
  const float* whh0 = (const float*)d_in[2];
  const float* bih0 = (const float*)d_in[3];
  const float* bhh0 = (const float*)d_in[4];
  const float* wih1 = (const float*)d_in[5];
  const float* whh1 = (const float*)d_in[6];
  const float* bih1 = (const float*)d_in[7];
  const float* bhh1 = (const float*)d_in[8];
  const float* wih2 = (const float*)d_in[9];
  const float* whh2 = (const float*)d_in[10];
  const float* bih2 = (const float*)d_in[11];
  const float* bhh2 = (const float*)d_in[12];
  const float* wlin = (const float*)d_in[13];
  const float* blin = (const float*)d_in[14];

  float* gi0 = (float*)d_ws;                       // [T,24]  6 MB
  float* h2  = gi0 + (size_t)T_SEQ * G;            // [T,8]   2 MB
  float* out = (float*)d_out;                      // [T,32]

  const int mblocks = T_SEQ / 16 / 8;              // 512 blocks, 8 wave-tiles each

  gi0_wmma<<<mblocks, 256, 0, stream>>>(x, wih0, bih0, gi0);
  gru_scan3<<<1, 96, 0, stream>>>(gi0, whh0, bhh0,
                                  wih1, whh1, bih1, bhh1,
                                  wih2, whh2, bih2, bhh2, h2);
  out_wmma<<<mblocks, 256, 0, stream>>>(h2, wlin, blin, out);
}